// RetroProbe_16681652977789
// MI455X (gfx1250) — compile-verified
//
#include <hip/hip_runtime.h>
#include <hip/hip_bf16.h>
#include <math.h>

// Problem constants (from reference): B, L, E, D, NL
#define Bb  16
#define Ll  256
#define Ee  768
#define Dd  128
#define NLl 48

typedef float v2f __attribute__((ext_vector_type(2)));
typedef float v8f __attribute__((ext_vector_type(8)));

// CDNA5 fp32 WMMA: D(16x16,f32) = A(16x4,f32) x B(4x16,f32) + C
__device__ __forceinline__ v8f wmma_k4(v2f a, v2f b, v8f c) {
    return __builtin_amdgcn_wmma_f32_16x16x4_f32(
        /*neg_a=*/false, a, /*neg_b=*/false, b,
        /*c_mod=*/(short)0, c, /*reuse_a=*/false, /*reuse_b=*/false);
}

// ---------------------------------------------------------------------------
// Kernel 1: dep = emb0 (4096x768) @ W_arc^T (768x128). One wave per 16x16 tile.
// grid = (4096/16) * (128/16) = 2048 waves.
// ---------------------------------------------------------------------------
__global__ __launch_bounds__(32) void k_dep_gemm(const float* __restrict__ emb0,
                                                 const float* __restrict__ W_arc,
                                                 float* __restrict__ dep) {
    const int tile = blockIdx.x;
    const int mt = tile >> 3;           // 256 row tiles
    const int nt = tile & 7;            // 8 col tiles
    const int lane = threadIdx.x;
    const int half = lane >> 4;         // 0: K=0..1 / 1: K=2..3
    const int lr   = lane & 15;

    const float* arow = emb0  + (size_t)(mt * 16 + lr) * Ee + 2 * half;
    const float* brow = W_arc + (size_t)(nt * 16 + lr) * Ee + 2 * half; // B[k][n] = W_arc[n][k]

    v8f acc = {};
    for (int k = 0; k < Ee; k += 4) {
        v2f a = *(const v2f*)(arow + k);
        v2f b = *(const v2f*)(brow + k);
        acc = wmma_k4(a, b, acc);
    }
    const int row0 = mt * 16, col0 = nt * 16;
#pragma unroll
    for (int v = 0; v < 8; ++v)
        dep[(size_t)(row0 + v + 8 * half) * Dd + col0 + lr] = acc[v];
}

// ---------------------------------------------------------------------------
// Kernel 2: norms[r] = ||dep[r]||^2, r in [0,4096)
// ---------------------------------------------------------------------------
__global__ __launch_bounds__(128) void k_norms(const float* __restrict__ dep,
                                               float* __restrict__ norms) {
    const int row = blockIdx.x * 128 + threadIdx.x;   // 32 blocks x 128
    const float4* p = (const float4*)(dep + (size_t)row * Dd);
    float s = 0.f;
#pragma unroll
    for (int i = 0; i < Dd / 4; ++i) {
        float4 v = p[i];
        s += v.x * v.x + v.y * v.y + v.z * v.z + v.w * v.w;
    }
    norms[row] = s;
}

// ---------------------------------------------------------------------------
// Kernel 3: distances[b,i,j] = n_i + n_j - 2 * dep_i . dep_j (per-batch Gram
// GEMM via WMMA). grid = 16 batches * 16*16 tiles = 4096 waves.
// ---------------------------------------------------------------------------
__global__ __launch_bounds__(32) void k_dist(const float* __restrict__ dep,
                                             const float* __restrict__ norms,
                                             float* __restrict__ dist) {
    const int tile = blockIdx.x;
    const int b  = tile >> 8;
    const int t  = tile & 255;
    const int it = t >> 4, jt = t & 15;
    const int lane = threadIdx.x;
    const int half = lane >> 4, lr = lane & 15;

    const float* G = dep + (size_t)b * Ll * Dd;
    const float* arow = G + (size_t)(it * 16 + lr) * Dd + 2 * half;
    const float* brow = G + (size_t)(jt * 16 + lr) * Dd + 2 * half;  // B = G^T

    v8f acc = {};
    for (int k = 0; k < Dd; k += 4) {
        v2f a  = *(const v2f*)(arow + k);
        v2f bb = *(const v2f*)(brow + k);
        acc = wmma_k4(a, bb, acc);
    }
    const float* nb = norms + (size_t)b * Ll;
    const float nj = nb[jt * 16 + lr];
    float* out = dist + (size_t)b * Ll * Ll;
#pragma unroll
    for (int v = 0; v < 8; ++v) {
        const int i = it * 16 + v + 8 * half;
        out[(size_t)i * Ll + jt * 16 + lr] = nb[i] + nj - 2.0f * acc[v];
    }
}

// ---------------------------------------------------------------------------
// Kernel 4: par = emb1 @ W_lbl[:, :E]^T ; chi = emb1 @ W_lbl[:, E:]^T
// N=48 -> 3 tiles; grid = 256 row-tiles * 3 col-tiles * 2 passes = 1536 waves.
// ---------------------------------------------------------------------------
__global__ __launch_bounds__(32) void k_parchi(const float* __restrict__ emb1,
                                               const float* __restrict__ W_lbl,
                                               float* __restrict__ par,
                                               float* __restrict__ chi) {
    const int p  = blockIdx.x & 1;          // 0 = parent, 1 = child
    const int t  = blockIdx.x >> 1;
    const int mt = t / 3, nt = t % 3;
    const int lane = threadIdx.x;
    const int half = lane >> 4, lr = lane & 15;

    const float* arow = emb1 + (size_t)(mt * 16 + lr) * Ee + 2 * half;
    const float* brow = W_lbl + (size_t)(nt * 16 + lr) * (2 * Ee) + p * Ee + 2 * half;

    v8f acc = {};
    for (int k = 0; k < Ee; k += 4) {
        v2f a = *(const v2f*)(arow + k);
        v2f b = *(const v2f*)(brow + k);
        acc = wmma_k4(a, b, acc);
    }
    float* out = p ? chi : par;
    const int row0 = mt * 16, col0 = nt * 16;
#pragma unroll
    for (int v = 0; v < 8; ++v)
        out[(size_t)(row0 + v + 8 * half) * NLl + col0 + lr] = acc[v];
}

// ---------------------------------------------------------------------------
// Kernel 5: parent_probs[row,:] = softmax(-distances[row,:]), one wave32/row,
// 8 elements per lane, shuffle-tree reductions.
// ---------------------------------------------------------------------------
__global__ __launch_bounds__(32) void k_softmax(const float* __restrict__ dist,
                                                float* __restrict__ probs) {
    const int row  = blockIdx.x;                    // 4096 rows
    const int lane = threadIdx.x;
    const float* d = dist + (size_t)row * Ll;
    float* pr      = probs + (size_t)row * Ll;

    float x[8];
    float m = -__builtin_inff();
#pragma unroll
    for (int i = 0; i < 8; ++i) {
        x[i] = -d[lane + 32 * i];
        m = fmaxf(m, x[i]);
    }
    for (int off = 16; off > 0; off >>= 1) m = fmaxf(m, __shfl_xor(m, off, 32));

    float s = 0.f;
#pragma unroll
    for (int i = 0; i < 8; ++i) { x[i] = expf(x[i] - m); s += x[i]; }
    for (int off = 16; off > 0; off >>= 1) s += __shfl_xor(s, off, 32);

    const float inv = 1.0f / s;
#pragma unroll
    for (int i = 0; i < 8; ++i) pr[lane + 32 * i] = x[i] * inv;
}

// ---------------------------------------------------------------------------
// Kernel 6: label_logits[b,j,n] = sum_k { valid ? logaddexp(chi+par, pp) : pp }
// One block per (b,j); probs/att rows staged in LDS; lanes own label n.
// ---------------------------------------------------------------------------
__global__ __launch_bounds__(64) void k_labels(const float* __restrict__ par,
                                               const float* __restrict__ chi,
                                               const float* __restrict__ probs,
                                               const int* __restrict__ att,
                                               float* __restrict__ out) {
    const int bj = blockIdx.x;       // 4096
    const int b  = bj >> 8, j = bj & 255;
    const int n  = threadIdx.x;

    __shared__ float s_pp[Ll];
    __shared__ int   s_att[Ll];
    for (int k = threadIdx.x; k < Ll; k += 64) {
        s_pp[k]  = probs[(size_t)bj * Ll + k];
        s_att[k] = att[b * Ll + k];
    }
    __syncthreads();

    if (n < NLl) {
        const float c  = chi[(size_t)bj * NLl + n];
        const int   aj = s_att[j];
        float acc = 0.f;
        for (int k = 0; k < Ll; ++k) {
            const float pp = s_pp[k];
            if (aj && s_att[k] && (k != j)) {
                const float xv = c + par[((size_t)b * Ll + k) * NLl + n];
                const float mx = fmaxf(xv, pp);
                const float dd = fminf(xv, pp) - mx;     // <= 0, finite
                acc += mx + log1pf(expf(dd));            // stable logaddexp
            } else {
                acc += pp;                               // logaddexp(-inf, pp)
            }
        }
        out[(size_t)bj * NLl + n] = acc;
    }
}

// ---------------------------------------------------------------------------
extern "C" void kernel_launch(void* const* d_in, const int* in_sizes, int n_in,
                              void* d_out, int out_size, void* d_ws, size_t ws_size,
                              hipStream_t stream) {
    (void)in_sizes; (void)n_in; (void)out_size; (void)ws_size;

    const float* emb0  = (const float*)d_in[0];
    const float* emb1  = (const float*)d_in[1];
    const int*   att   = (const int*)  d_in[2];
    const float* W_arc = (const float*)d_in[3];
    const float* W_lbl = (const float*)d_in[4];

    // Output tuple, concatenated flat: dep | distances | label_logits
    float* dep  = (float*)d_out;                       // 16*256*128
    float* dist = dep  + (size_t)Bb * Ll * Dd;         // 16*256*256
    float* lbl  = dist + (size_t)Bb * Ll * Ll;         // 16*256*48

    // Workspace carve-up (floats): probs | par | chi | norms  (~5.8 MB)
    float* ws    = (float*)d_ws;
    float* probs = ws;                                 // 16*256*256
    float* par   = probs + (size_t)Bb * Ll * Ll;       // 16*256*48
    float* chi   = par   + (size_t)Bb * Ll * NLl;      // 16*256*48
    float* norms = chi   + (size_t)Bb * Ll * NLl;      // 16*256

    k_dep_gemm<<<(Bb * Ll / 16) * (Dd / 16), 32, 0, stream>>>(emb0, W_arc, dep);
    k_norms   <<<(Bb * Ll) / 128, 128, 0, stream>>>(dep, norms);
    k_dist    <<<Bb * (Ll / 16) * (Ll / 16), 32, 0, stream>>>(dep, norms, dist);
    k_parchi  <<<(Bb * Ll / 16) * 3 * 2, 32, 0, stream>>>(emb1, W_lbl, par, chi);
    k_softmax <<<Bb * Ll, 32, 0, stream>>>(dist, probs);
    k_labels  <<<Bb * Ll, 64, 0, stream>>>(par, chi, probs, att, lbl);
}